// APPNP_5772436045966
// MI455X (gfx1250) — compile-verified
//
#include <hip/hip_runtime.h>
#include <math.h>

// Problem constants (from reference)
#define NN    100000
#define EE    1600000
#define DIN   512
#define DH    256
#define DOUT  40
#define KSTEPS 10
#define ALPHA_C 0.5f
#define EPS_C   1e-5f

typedef __attribute__((ext_vector_type(2))) float v2f;
typedef __attribute__((ext_vector_type(8))) float v8f;

// V_WMMA_F32_16X16X4_F32 : D[16x16] = A[16x4] * B[4x16] + C, fp32 matrix pipe.
// 8-arg form: (neg_a, A, neg_b, B, c_mod, C, reuse_a, reuse_b)
__device__ __forceinline__ v8f wmma4(v2f a, v2f b, v8f c) {
  return __builtin_amdgcn_wmma_f32_16x16x4_f32(false, a, false, b, (short)0, c,
                                               false, false);
}

// ---------------------------------------------------------------------------
// Fused MLP: per 16-row tile (one wave32 per workgroup):
//   h1 = relu(bn1(x @ W1 + b1))   (LDS)
//   h2 = relu(bn2(h1 @ W2 + b2))  (LDS)
//   h0 = h2 @ W3 + b3             (global, [N, 40])
// A-matrix per-lane layout (16x4 fp32): lane m=L&15, K pair = k0 + 2*(L>>4).
// B-matrix per-lane layout (4x16 fp32): col n = L&15, K pair = k0 + 2*(L>>4).
// C/D layout: VGPR j -> M = j + 8*(L>>4), N = L&15.
// ---------------------------------------------------------------------------
__global__ __launch_bounds__(32) void mlp_fused_kernel(
    const float* __restrict__ x,
    const float* __restrict__ W1, const float* __restrict__ b1,
    const float* __restrict__ g1, const float* __restrict__ be1,
    const float* __restrict__ m1, const float* __restrict__ v1,
    const float* __restrict__ W2, const float* __restrict__ b2,
    const float* __restrict__ g2, const float* __restrict__ be2,
    const float* __restrict__ m2, const float* __restrict__ v2,
    const float* __restrict__ W3, const float* __restrict__ b3,
    float* __restrict__ h0out)
{
  __shared__ float h1s[16 * DH];   // 16 KB
  __shared__ float h2s[16 * DH];   // 16 KB

  const int lane  = threadIdx.x;        // 0..31
  const int mrow  = lane & 15;          // row within tile for A loads
  const int nlane = lane & 15;          // column within 16-wide N tile
  const int hi    = lane >> 4;          // 0 or 1
  const int khalf = hi * 2;             // K sub-pair offset
  const int r0    = blockIdx.x * 16;    // N = 100000 = 6250 * 16, exact

  // ---------------- Stage 1: x @ W1 -> h1 (LDS) ----------------
  const float* xrow = x + (size_t)(r0 + mrow) * DIN;
  for (int nt = 0; nt < DH / 16; ++nt) {
    const int n = nt * 16 + nlane;
    v8f acc = {0.f, 0.f, 0.f, 0.f, 0.f, 0.f, 0.f, 0.f};
    for (int k0 = 0; k0 < DIN; k0 += 4) {
      const int kk = k0 + khalf;
      v2f a = *(const v2f*)(xrow + kk);            // 8B aligned (kk even)
      v2f b;
      b.x = W1[(size_t)kk * DH + n];
      b.y = W1[(size_t)(kk + 1) * DH + n];
      acc = wmma4(a, b, acc);
    }
    // epilogue: bn1(acc + b1) then relu ; params depend only on column n
    const float s  = g1[n] * rsqrtf(v1[n] + EPS_C);
    const float sh = be1[n] - m1[n] * s;
    const float bb = b1[n];
#pragma unroll
    for (int j = 0; j < 8; ++j) {
      const int mm = j + hi * 8;
      float val = (acc[j] + bb) * s + sh;
      h1s[mm * DH + n] = fmaxf(val, 0.f);
    }
  }
  __syncthreads();

  // ---------------- Stage 2: h1 @ W2 -> h2 (LDS) ----------------
  for (int nt = 0; nt < DH / 16; ++nt) {
    const int n = nt * 16 + nlane;
    v8f acc = {0.f, 0.f, 0.f, 0.f, 0.f, 0.f, 0.f, 0.f};
    for (int k0 = 0; k0 < DH; k0 += 4) {
      const int kk = k0 + khalf;
      v2f a = *(const v2f*)(&h1s[mrow * DH + kk]); // ds_load_b64
      v2f b;
      b.x = W2[(size_t)kk * DH + n];
      b.y = W2[(size_t)(kk + 1) * DH + n];
      acc = wmma4(a, b, acc);
    }
    const float s  = g2[n] * rsqrtf(v2[n] + EPS_C);
    const float sh = be2[n] - m2[n] * s;
    const float bb = b2[n];
#pragma unroll
    for (int j = 0; j < 8; ++j) {
      const int mm = j + hi * 8;
      float val = (acc[j] + bb) * s + sh;
      h2s[mm * DH + n] = fmaxf(val, 0.f);
    }
  }
  __syncthreads();

  // ---------------- Stage 3: h2 @ W3 + b3 -> h0out [N,40] ----------------
  for (int nt = 0; nt < 3; ++nt) {                 // 48 cols, last 8 masked
    const int n = nt * 16 + nlane;
    const bool valid = (n < DOUT);
    v8f acc = {0.f, 0.f, 0.f, 0.f, 0.f, 0.f, 0.f, 0.f};
    for (int k0 = 0; k0 < DH; k0 += 4) {
      const int kk = k0 + khalf;
      v2f a = *(const v2f*)(&h2s[mrow * DH + kk]);
      v2f b;
      b.x = valid ? W3[(size_t)kk * DOUT + n] : 0.f;
      b.y = valid ? W3[(size_t)(kk + 1) * DOUT + n] : 0.f;
      acc = wmma4(a, b, acc);
    }
    const float bb = valid ? b3[n] : 0.f;
#pragma unroll
    for (int j = 0; j < 8; ++j) {
      const int mm = j + hi * 8;
      if (valid) h0out[(size_t)(r0 + mm) * DOUT + n] = acc[j] + bb;
    }
  }
}

// ---------------------------------------------------------------------------
// Propagation: next = ALPHA*ori  then  next[row] += (1-ALPHA)*w*cur[col]
// ---------------------------------------------------------------------------
__global__ void scale_init_kernel(const float* __restrict__ ori,
                                  float* __restrict__ next, int count) {
  int i = blockIdx.x * blockDim.x + threadIdx.x;
  if (i < count) next[i] = ALPHA_C * ori[i];
}

__global__ void edge_scatter_kernel(const int* __restrict__ row,
                                    const int* __restrict__ col,
                                    const float* __restrict__ w,
                                    const float* __restrict__ cur,
                                    float* __restrict__ next, int e_total) {
  int gid = blockIdx.x * blockDim.x + threadIdx.x;
  int e = gid >> 3;                  // 8 lanes per edge
  if (e >= e_total) return;
  int l8 = gid & 7;
  int r = row[e];
  int c = col[e];
  float wv = (1.0f - ALPHA_C) * w[e];
  const float* src = cur + (size_t)c * DOUT;
  float* dst = next + (size_t)r * DOUT;
#pragma unroll
  for (int i = 0; i < 5; ++i) {      // 8 lanes * 5 = 40 features
    int j = l8 + 8 * i;
    unsafeAtomicAdd(dst + j, wv * src[j]);   // global_atomic_add_f32
  }
}

// ---------------------------------------------------------------------------
// Row-wise log_softmax over 40 classes
// ---------------------------------------------------------------------------
__global__ void log_softmax_kernel(const float* __restrict__ in,
                                   float* __restrict__ out, int n) {
  int i = blockIdx.x * blockDim.x + threadIdx.x;
  if (i >= n) return;
  const float* p = in + (size_t)i * DOUT;
  float mx = p[0];
#pragma unroll
  for (int j = 1; j < DOUT; ++j) mx = fmaxf(mx, p[j]);
  float s = 0.f;
#pragma unroll
  for (int j = 0; j < DOUT; ++j) s += expf(p[j] - mx);
  float ls = logf(s);
  float* q = out + (size_t)i * DOUT;
#pragma unroll
  for (int j = 0; j < DOUT; ++j) q[j] = p[j] - mx - ls;
}

extern "C" void kernel_launch(void* const* d_in, const int* in_sizes, int n_in,
                              void* d_out, int out_size, void* d_ws,
                              size_t ws_size, hipStream_t stream) {
  (void)in_sizes; (void)n_in; (void)out_size; (void)ws_size;

  const float* x   = (const float*)d_in[0];
  const int*   row = (const int*)  d_in[1];
  const int*   col = (const int*)  d_in[2];
  const float* w   = (const float*)d_in[3];
  const float* W1  = (const float*)d_in[4];
  const float* b1  = (const float*)d_in[5];
  const float* g1  = (const float*)d_in[6];
  const float* be1 = (const float*)d_in[7];
  const float* m1  = (const float*)d_in[8];
  const float* v1  = (const float*)d_in[9];
  const float* W2  = (const float*)d_in[10];
  const float* b2  = (const float*)d_in[11];
  const float* g2  = (const float*)d_in[12];
  const float* be2 = (const float*)d_in[13];
  const float* m2  = (const float*)d_in[14];
  const float* v2  = (const float*)d_in[15];
  const float* W3  = (const float*)d_in[16];
  const float* b3  = (const float*)d_in[17];
  float* out = (float*)d_out;

  // Workspace layout (floats): [ori: N*40][bufA: N*40][bufB: N*40] = 48 MB
  float* ori  = (float*)d_ws;
  float* bufA = ori  + (size_t)NN * DOUT;
  float* bufB = bufA + (size_t)NN * DOUT;

  // 1) fused MLP -> ori
  mlp_fused_kernel<<<NN / 16, 32, 0, stream>>>(
      x, W1, b1, g1, be1, m1, v1, W2, b2, g2, be2, m2, v2, W3, b3, ori);

  // 2) K rounds of APPNP propagation (ping-pong)
  const int cnt = NN * DOUT;
  const float* cur = ori;
  for (int it = 0; it < KSTEPS; ++it) {
    float* next = (it & 1) ? bufB : bufA;
    scale_init_kernel<<<(cnt + 255) / 256, 256, 0, stream>>>(ori, next, cnt);
    edge_scatter_kernel<<<((size_t)EE * 8 + 255) / 256, 256, 0, stream>>>(
        row, col, w, cur, next, EE);
    cur = next;
  }

  // 3) log_softmax -> d_out
  log_softmax_kernel<<<(NN + 255) / 256, 256, 0, stream>>>(cur, out, NN);
}